// PiecewiseLinearEmbedding_60309930770840
// MI455X (gfx1250) — compile-verified
//
#include <hip/hip_runtime.h>
#include <stdint.h>

// Problem constants (from reference): B=8192, F=256, E=49, D=64
#define PLE_B 8192
#define PLE_F 256
#define PLE_E 49
#define PLE_D 64
#define NB    256                    // batches handled per block
#define BDIM  256                    // threads per block (8 waves of 32)
#define EMB_F_FLOATS (PLE_E * PLE_D) // 3136 floats = 12544 bytes per feature
#define EMB_CHUNKS   (EMB_F_FLOATS * 4 / 16) // 784 x 16-byte chunks (3*256 + 16)

typedef __attribute__((ext_vector_type(4))) float float4v;
typedef __attribute__((ext_vector_type(2))) float float2v;

__device__ __forceinline__ void async_copy_b128(uint32_t lds_off, uint32_t glob_off,
                                                uint64_t sbase) {
    // GVS mode: mem_addr = SGPR64 base + VGPR32 offset; first operand = LDS byte addr.
    asm volatile("global_load_async_to_lds_b128 %0, %1, %2"
                 :: "v"(lds_off), "v"(glob_off), "s"(sbase)
                 : "memory");
}

__global__ __launch_bounds__(BDIM) void ple_kernel(
    const float* __restrict__ x,      // (B, F)
    const float* __restrict__ edges,  // (F, E)
    const float* __restrict__ emb,    // (F, E, D)
    float* __restrict__ out)          // (B, F, D)
{
    __shared__ alignas(16) float   sEmb[EMB_F_FLOATS]; // emb[f] slice (12.5 KB)
    __shared__ float               sEdge[PLE_E];
    __shared__ alignas(8) float2v  sTI[NB];            // {t, bitcast(idx)} per batch row

    const int tid = threadIdx.x;
    const int f   = blockIdx.x & (PLE_F - 1);          // adjacent blocks -> adjacent f
    const int b0  = (blockIdx.x >> 8) * NB;

    // ---- Stage emb[f] into LDS with gfx1250 async global->LDS copies (ASYNCcnt).
    // Straight-line issue: 3 full-width rounds + guarded 16-chunk tail.
    {
        const uint32_t ldsBase = (uint32_t)(uintptr_t)&sEmb[0]; // low 32 bits of the
                                                                // generic LDS ptr ==
                                                                // LDS byte offset
        const uint64_t gbase =
            (uint64_t)(uintptr_t)(emb + (size_t)f * EMB_F_FLOATS); // uniform -> SGPRs
        #pragma unroll
        for (int i = 0; i < 3; ++i) {
            const uint32_t c = (uint32_t)tid + (uint32_t)i * BDIM;
            async_copy_b128(ldsBase + c * 16u, c * 16u, gbase);
        }
        if (tid < EMB_CHUNKS - 3 * BDIM) { // 16-chunk tail
            const uint32_t c = (uint32_t)tid + 3u * BDIM;
            async_copy_b128(ldsBase + c * 16u, c * 16u, gbase);
        }
    }

    // ---- Edges to LDS (tiny, synchronous).
    if (tid < PLE_E) sEdge[tid] = edges[f * PLE_E + tid];
    __syncthreads(); // phase 1 only needs sEdge (async emb writes still in flight)

    // ---- Phase 1: one thread per batch row -> searchsorted + interpolation weight.
    {
        const float v = x[(size_t)(b0 + tid) * PLE_F + f];
        // searchsorted(inner, v, 'left') over inner = sEdge[1..47]: smallest i with
        // inner[i] >= v, result in [0, 47]
        int lo = 0, hi = PLE_E - 2; // 47
        while (lo < hi) {
            int mid = (lo + hi) >> 1;
            if (sEdge[mid + 1] < v) lo = mid + 1; else hi = mid;
        }
        const int   idx = lo;
        const float le  = sEdge[idx];
        const float re  = sEdge[idx + 1];
        float den = re - le;
        den = (fabsf(den) < 1e-8f) ? 1.0f : den;
        float t = (v - le) / den;
        t = fminf(fmaxf(t, 0.0f), 1.0f);
        float2v ti;
        ti.x = t;
        ti.y = __int_as_float(idx);
        sTI[tid] = ti;
    }

    // Drain this wave's async copies, then barrier: emb + t/idx now visible to all.
    asm volatile("s_wait_asynccnt 0" ::: "memory");
    __syncthreads();

    // ---- Phase 2: stream the lerped rows. 16 lanes cover one 256-byte output row
    // as contiguous b128 stores; write-once output uses non-temporal hints.
    const int d4   = tid & 15;  // which float4 of the D=64 row
    const int bsub = tid >> 4;  // 0..15
    float* outBase = out + ((size_t)b0 * PLE_F + f) * PLE_D;

    #pragma unroll
    for (int it = 0; it < NB / 16; ++it) {
        const int     bi  = it * 16 + bsub;
        const float2v ti  = sTI[bi];           // one ds_load_b64, 16-lane broadcast
        const float   t   = ti.x;
        const int     idx = __float_as_int(ti.y);
        const float4v L = *(const float4v*)(sEmb + idx * PLE_D + d4 * 4);
        const float4v R = *(const float4v*)(sEmb + (idx + 1) * PLE_D + d4 * 4);
        const float4v r = (1.0f - t) * L + t * R;   // exact reference expression
        float4v* dst = (float4v*)(outBase + (size_t)bi * PLE_F * PLE_D + d4 * 4);
        __builtin_nontemporal_store(r, dst);
    }
}

extern "C" void kernel_launch(void* const* d_in, const int* in_sizes, int n_in,
                              void* d_out, int out_size, void* d_ws, size_t ws_size,
                              hipStream_t stream) {
    const float* x     = (const float*)d_in[0];
    const float* edges = (const float*)d_in[1];
    const float* emb   = (const float*)d_in[2];
    float* out = (float*)d_out;

    const int nblocks = (PLE_B / NB) * PLE_F; // 8192
    ple_kernel<<<dim3(nblocks), dim3(BDIM), 0, stream>>>(x, edges, emb, out);
}